// WongWangMultiClassDecision_14826227106273
// MI455X (gfx1250) — compile-verified
//
#include <hip/hip_runtime.h>
#include <cstdint>

#define BLOCK   64
#define KDEPTH  8          // async ring depth (steps in flight)
#define T_STEPS 500
#define NBATCH  16384

// Wong-Wang constants (folded at compile time)
#define OU_DECAY   0.006737946999085467f   // exp(-DT/TAU_AMPA) = exp(-5)
#define OU_SCALE   0.70709073f             // sqrt((1-exp(-10))/2)

typedef __attribute__((address_space(3))) unsigned char lds_byte_t;

// Issue one per-lane 16B async copy global -> LDS (tracked by ASYNCcnt).
__device__ __forceinline__ void async_copy16(uint32_t lds_off, uint64_t gaddr) {
    asm volatile("global_load_async_to_lds_b128 %0, %1, off"
                 :: "v"(lds_off), "v"(gaddr) : "memory");
}

#define WAIT_ASYNCCNT(n) asm volatile("s_wait_asynccnt " #n ::: "memory")

__global__ __launch_bounds__(BLOCK)
void wong_wang_decision_kernel(const float* __restrict__ input_signal,
                               const float* __restrict__ noise_init,
                               const float* __restrict__ noise_seq,
                               const float* __restrict__ Jmat,
                               const float* __restrict__ p_Jext,
                               const float* __restrict__ p_I0,
                               const float* __restrict__ p_na,
                               const float* __restrict__ p_th,
                               float* __restrict__ out)
{
    __shared__ float4 ring[KDEPTH * BLOCK];

    const int tid = threadIdx.x;
    const int gid = blockIdx.x * BLOCK + tid;

    const float Jext = p_Jext[0];
    const float I0   = p_I0[0];
    const float na   = p_na[0];
    const float th   = p_th[0];
    const float nsc  = na * OU_SCALE;      // noise_ampa * ou_scale

    // J is 4x4, uniform across threads -> scalar loads
    float Jm[16];
#pragma unroll
    for (int i = 0; i < 16; ++i) Jm[i] = Jmat[i];

    const float4 in4 = ((const float4*)input_signal)[gid];
    const float4 nz0 = ((const float4*)noise_init)[gid];

    float Ic[4]  = { I0 + Jext * in4.x, I0 + Jext * in4.y,
                     I0 + Jext * in4.z, I0 + Jext * in4.w };
    float s[4]   = { 0.1f, 0.1f, 0.1f, 0.1f };
    float In[4]  = { nz0.x * na, nz0.y * na, nz0.z * na, nz0.w * na };
    float dec[4] = { (float)(T_STEPS - 1), (float)(T_STEPS - 1),
                     (float)(T_STEPS - 1), (float)(T_STEPS - 1) };
    bool  done[4] = { false, false, false, false };

    // ---- async ring setup --------------------------------------------------
    // LDS byte offset of this thread's slot 0 (addrspace(3) ptrtoint).
    const uint32_t lds_base =
        (uint32_t)(uintptr_t)((lds_byte_t*)ring) + (uint32_t)tid * 16u;

    uint64_t ga = (uint64_t)(uintptr_t)noise_seq + (uint64_t)gid * 16u;
    const uint64_t gstride = (uint64_t)NBATCH * 16u;   // bytes per timestep

#pragma unroll
    for (int k = 0; k < KDEPTH; ++k) {                 // prologue: fill ring
        async_copy16(lds_base + (uint32_t)k * (BLOCK * 16u), ga);
        ga += gstride;
    }

    // ---- one Wong-Wang step -----------------------------------------------
    auto step = [&](const float4 nz, float tval) {
        const float noise[4] = { nz.x, nz.y, nz.z, nz.w };
        float x[4];
#pragma unroll
        for (int c = 0; c < 4; ++c) {
            x[c] = Ic[c] + In[c];
            x[c] = fmaf(s[0], Jm[0 * 4 + c], x[c]);
            x[c] = fmaf(s[1], Jm[1 * 4 + c], x[c]);
            x[c] = fmaf(s[2], Jm[2 * 4 + c], x[c]);
            x[c] = fmaf(s[3], Jm[3 * 4 + c], x[c]);
        }
#pragma unroll
        for (int c = 0; c < 4; ++c) {
            const float u   = fmaf(270.0f, x[c], -108.0f);
            const float e   = __expf(-0.154f * u);
            const float den = (1.0f + 1e-6f) - e;
            float H = u * __builtin_amdgcn_rcpf(den);
            H = fmaxf(H, 0.0f);
            // dsdt = -s/100 + (1-s)*H*0.000641 ; s += dsdt*10
            const float ds = fmaf((1.0f - s[c]) * H, 0.000641f, -s[c] * 0.01f);
            In[c] = fmaf(In[c], OU_DECAY, nsc * noise[c]);
            s[c]  = fmaf(ds, 10.0f, s[c]);
            if (!done[c] && s[c] > th) { dec[c] = tval; done[c] = true; }
        }
    };

    // ---- main loop: consume slot t, refill with step t+K -------------------
    int t = 0;
    for (; t < T_STEPS - KDEPTH; ++t) {
        WAIT_ASYNCCNT(7);                               // load for step t done
        const uint32_t slot = (uint32_t)(t & (KDEPTH - 1));
        const float4 nz = ring[slot * BLOCK + tid];
        step(nz, (float)t);
        async_copy16(lds_base + slot * (BLOCK * 16u), ga);
        ga += gstride;
    }
    WAIT_ASYNCCNT(0);                                   // drain ring
    for (; t < T_STEPS; ++t) {
        const float4 nz = ring[(uint32_t)(t & (KDEPTH - 1)) * BLOCK + tid];
        step(nz, (float)t);
    }

    // decision time = step_index * DT / 1000 = idx * 0.01
    float4 o = { dec[0] * 0.01f, dec[1] * 0.01f, dec[2] * 0.01f, dec[3] * 0.01f };
    ((float4*)out)[gid] = o;
}

extern "C" void kernel_launch(void* const* d_in, const int* in_sizes, int n_in,
                              void* d_out, int out_size, void* d_ws, size_t ws_size,
                              hipStream_t stream) {
    const float* input_signal = (const float*)d_in[0];
    const float* noise_init   = (const float*)d_in[1];
    const float* noise_seq    = (const float*)d_in[2];
    const float* Jmat         = (const float*)d_in[3];
    const float* Jext         = (const float*)d_in[4];
    const float* I0           = (const float*)d_in[5];
    const float* na           = (const float*)d_in[6];
    const float* th           = (const float*)d_in[7];

    dim3 grid(NBATCH / BLOCK);
    dim3 block(BLOCK);
    wong_wang_decision_kernel<<<grid, block, 0, stream>>>(
        input_signal, noise_init, noise_seq, Jmat, Jext, I0, na, th,
        (float*)d_out);
}